// Watson_29420525977975
// MI455X (gfx1250) — compile-verified
//
#include <hip/hip_runtime.h>
#include <hip/hip_bf16.h>
#include <math.h>

// ---------------------------------------------------------------------------
// Watson log-pdf:
//   out[i] = logC + softplus(kappa) * (X[i,:] . mu_unit)^2
// Bulk = 1.024 GB stream of X -> HBM-bound (~44 us @ 23.3 TB/s).
// GEMV done in full fp32 via V_WMMA_F32_16X16X4_F32 (16 rows / wave-tile).
// X is read exactly once (5x the 192MB L2) -> non-temporal load hints so the
// stream does not thrash the caches holding the reused mu_unit/ws data.
// ---------------------------------------------------------------------------

typedef __attribute__((ext_vector_type(2))) float v2f;
typedef __attribute__((ext_vector_type(8))) float v8f;

#define P_DIM 128           // feature dim (reference: p = 128)
#define KUMMER_TERMS 10000

// ---------------------------------------------------------------------------
// Kernel 1: scalar prework (single 256-thread block).
//   ws[0] = logC, ws[1] = softplus(kappa), ws[2..2+p-1] = mu_unit
// ---------------------------------------------------------------------------
__global__ __launch_bounds__(256) void watson_prep_kernel(
    const float* __restrict__ mu, const float* __restrict__ kappa,
    float* __restrict__ ws, int p) {
  __shared__ float red_m[256];
  __shared__ float red_s[256];
  __shared__ float bcast;
  const int t = threadIdx.x;

  // ---- ||mu||^2 reduction ----
  float ssq = 0.f;
  for (int i = t; i < p; i += 256) { float v = mu[i]; ssq += v * v; }
  red_m[t] = ssq;
  __syncthreads();
  for (int off = 128; off > 0; off >>= 1) {
    if (t < off) red_m[t] += red_m[t + off];
    __syncthreads();
  }
  if (t == 0) bcast = 1.0f / fmaxf(sqrtf(red_m[0]), 1e-12f);
  __syncthreads();
  const float inv_norm = bcast;
  for (int i = t; i < p; i += 256) ws[2 + i] = mu[i] * inv_norm;
  __syncthreads();

  // ---- 10000-term Kummer 1F1 logsumexp (online per-thread, LDS combine) ----
  const float kap    = kappa[0];
  const float logkap = logf(kap);
  const float a   = 0.5f;
  const float b   = 0.5f * (float)p;
  const float lga = lgammaf(a);
  const float lgb = lgammaf(b);
  float m = -INFINITY, s = 0.f;
  for (int n = t; n < KUMMER_TERMS; n += 256) {
    const float fn = (float)n;
    const float x = lgammaf(a + fn) + lgb - lga - lgammaf(b + fn)
                  + fn * logkap - lgammaf(fn + 1.f);
    if (x > m) { s = s * expf(m - x) + 1.f; m = x; }
    else       { s += expf(x - m); }
  }
  red_m[t] = m;
  red_s[t] = s;
  __syncthreads();
  for (int off = 128; off > 0; off >>= 1) {
    if (t < off) {
      const float m1 = red_m[t], s1 = red_s[t];
      const float m2 = red_m[t + off], s2 = red_s[t + off];
      const float mm = fmaxf(m1, m2);
      float ss = 0.f;
      if (m1 > -INFINITY) ss += s1 * expf(m1 - mm);
      if (m2 > -INFINITY) ss += s2 * expf(m2 - mm);
      red_m[t] = mm;
      red_s[t] = ss;
    }
    __syncthreads();
  }
  if (t == 0) {
    const float logKummer = red_m[0] + logf(red_s[0]);
    // log(2 * pi^(p/2)) = log 2 + (p/2) * log pi
    const float logc = 0.69314718055994530942f
                     + (0.5f * (float)p) * 1.14472988584940017414f;
    ws[0] = lgb - logc - logKummer;                       // logC
    ws[1] = (kap > 20.f) ? kap : log1pf(expf(kap));      // softplus(kappa)
  }
}

// ---------------------------------------------------------------------------
// Kernel 2: streamed GEMV + finalize via fp32 WMMA.
//
// Per wave-tile of 16 rows, per WMMA step j (K-base k0 = 4j), using the
// documented 32-bit operand layouts (lane-half splits K; VGPR index advances
// K within the half):
//   A vgpr v, lane L:  A[M = L&15][k0 + v + 2*(L>=16)]  -> one b64 NT load/lane
//   B vgpr v, lane L:  B[k0 + v + 2*(L>=16)][L&15]       = mu_unit[...] (bcast)
// D column N=0 then carries proj for rows M=0..7 (lane 0) / 8..15 (lane 16).
// A and B share the same K mapping, so the accumulated dot is exact.
// ---------------------------------------------------------------------------
__global__ __launch_bounds__(256) void watson_main_kernel(
    const float* __restrict__ X, const float* __restrict__ ws,
    float* __restrict__ out, int N) {
  const int lane  = threadIdx.x & 31;
  const int half  = lane >> 4;    // 0 -> K%4 in {0,1}; 1 -> K%4 in {2,3}
  const int mrow  = lane & 15;    // row within 16-row tile
  const int waveId = (blockIdx.x * (blockDim.x >> 5)) + (threadIdx.x >> 5);
  const int nWaves = gridDim.x * (blockDim.x >> 5);

  const float logC = ws[0];
  const float kpos = ws[1];
  const float* __restrict__ muU = ws + 2;

  // Preload replicated-B operands once per wave (32 x float2 in VGPRs).
  v2f bmat[32];
#pragma unroll
  for (int j = 0; j < 32; ++j) {
    bmat[j] = *(const v2f*)(muU + 4 * j + 2 * half);
  }

  const int nTiles = (N + 15) >> 4;
  for (int tile = waveId; tile < nTiles; tile += nWaves) {
    const int rowBase = tile << 4;
    int row = rowBase + mrow;
    if (row >= N) row = N - 1;  // clamp so EXEC stays all-ones for WMMA
    const float* __restrict__ xrow =
        X + (size_t)row * P_DIM + 2 * half;

    // Prefetch the first line of this lane's row in the next tile (GL2).
    __builtin_prefetch(xrow + (size_t)nWaves * 16 * P_DIM, 0, 0);

    v8f acc0 = {0.f, 0.f, 0.f, 0.f, 0.f, 0.f, 0.f, 0.f};
    v8f acc1 = {0.f, 0.f, 0.f, 0.f, 0.f, 0.f, 0.f, 0.f};
#pragma unroll
    for (int j = 0; j < 32; j += 2) {
      // Non-temporal: X is a read-once 1GB stream; do not pollute WGP$/L2.
      const v2f a0 = __builtin_nontemporal_load((const v2f*)(xrow + 4 * j));
      const v2f a1 = __builtin_nontemporal_load((const v2f*)(xrow + 4 * (j + 1)));
      acc0 = __builtin_amdgcn_wmma_f32_16x16x4_f32(
          false, a0, false, bmat[j],     (short)0, acc0, false, false);
      acc1 = __builtin_amdgcn_wmma_f32_16x16x4_f32(
          false, a1, false, bmat[j + 1], (short)0, acc1, false, false);
    }
    const v8f proj = acc0 + acc1;

    // Column N=0 lives in lanes 0 (M=0..7) and 16 (M=8..15).
    if (mrow == 0) {
      const int outBase = rowBase + 8 * half;
      if (outBase + 8 <= N) {
#pragma unroll
        for (int r = 0; r < 8; ++r) {
          const float pr = proj[r];
          __builtin_nontemporal_store(fmaf(kpos, pr * pr, logC),
                                      out + outBase + r);
        }
      } else {
#pragma unroll
        for (int r = 0; r < 8; ++r) {
          if (outBase + r < N) {
            const float pr = proj[r];
            __builtin_nontemporal_store(fmaf(kpos, pr * pr, logC),
                                        out + outBase + r);
          }
        }
      }
    }
  }
}

// ---------------------------------------------------------------------------
// Launch
//   d_in[0] = X [N, 128] f32, d_in[1] = mu [128] f32, d_in[2] = kappa [1] f32
//   d_out   = [N] f32
//   d_ws    : ws[0]=logC, ws[1]=softplus(kappa), ws[2..129]=mu_unit
// ---------------------------------------------------------------------------
extern "C" void kernel_launch(void* const* d_in, const int* in_sizes, int n_in,
                              void* d_out, int out_size, void* d_ws, size_t ws_size,
                              hipStream_t stream) {
  const float* X     = (const float*)d_in[0];
  const float* mu    = (const float*)d_in[1];
  const float* kappa = (const float*)d_in[2];
  float* out = (float*)d_out;
  float* ws  = (float*)d_ws;

  const int p = in_sizes[1];            // 128
  const int N = in_sizes[0] / p;        // 2,000,000

  watson_prep_kernel<<<1, 256, 0, stream>>>(mu, kappa, ws, p);

  const int nTiles        = (N + 15) >> 4;
  const int wavesPerBlock = 8;          // 256 threads, wave32
  int blocks = (nTiles + wavesPerBlock - 1) / wavesPerBlock;
  if (blocks > 16384) blocks = 16384;   // grid-stride covers the rest
  watson_main_kernel<<<blocks, 256, 0, stream>>>(X, ws, out, N);
}